// QueryAndGroup_5153960755815
// MI455X (gfx1250) — compile-verified
//
#include <hip/hip_runtime.h>
#include <hip/hip_bf16.h>

// PointNet++ ball-query + grouping for MI455X (gfx1250, wave32, WMMA).
// B=8, N=8192, S=2048, C=64, RADIUS=0.1, NSAMPLE=32. Output (B, 3+C, S, 32) f32.
//
// Distance tiles via V_WMMA_F32_16X16X4_F32:
//   A(16x4) row m  = (-2qx, -2qy, -2qz, 0)    (16 queries)
//   B(4x16) col n  = ( px,   py,   pz,  0)    (16 points)
//   C[m][n]        = |q_m|^2
//   D[m][n]        = |q_m|^2 - 2 q_m.p_n
// In-radius test:  D[m][n] <= r^2 - |p_n|^2  (per-COLUMN threshold; in the
// C/D layout each lane holds exactly one column == the point that lane loaded
// for B, so the threshold is a single per-lane scalar).
// xyz is staged to LDS byte-for-byte with GLOBAL_LOAD_ASYNC_TO_LDS_B128
// (inline asm; ASYNCcnt tracked, drained with s_wait_asynccnt 0).

#define BQ_B 8
#define BQ_N 8192
#define BQ_S 2048
#define BQ_C 64
#define BQ_NS 32
#define BQ_R2 0.01f
#define BQ_QPB 64           // queries per block (4 waves x 16)
#define BQ_THREADS 128
#define BQ_OUTC (3 + BQ_C)  // 67

typedef __attribute__((ext_vector_type(2))) float v2f;
typedef __attribute__((ext_vector_type(8))) float v8f;

__device__ __forceinline__ void async_copy_b128(const void* gsrc, void* ldst) {
    // vdst = VGPR with LDS byte offset (generic LDS addr low 32 bits == LDS offset),
    // vaddr = 64-bit global address (VGPR pair), no SADDR.
    asm volatile("global_load_async_to_lds_b128 %0, %1, off"
                 :: "v"((unsigned)(size_t)ldst), "v"((unsigned long long)(size_t)gsrc)
                 : "memory");
}

__device__ __forceinline__ void wait_async0() {
    asm volatile("s_wait_asynccnt 0x0" ::: "memory");
}

__global__ void ball_query_group_kernel(const float* __restrict__ xyz,      // (B,N,3)
                                        const float* __restrict__ new_xyz,  // (B,S,3)
                                        const float* __restrict__ features, // (B,C,N)
                                        float* __restrict__ out)            // (B,67,S,32)
{
    extern __shared__ char smem_raw[];
    float* raw3   = reinterpret_cast<float*>(smem_raw);                       // [N*3] 96KB, byte-copy of xyz[b]
    int*   idxlds = reinterpret_cast<int*>(smem_raw + sizeof(float) * BQ_N * 3); // [QPB][NS] 8KB

    const int blocks_per_batch = BQ_S / BQ_QPB;  // 32
    const int b      = blockIdx.x / blocks_per_batch;
    const int s_base = (blockIdx.x % blocks_per_batch) * BQ_QPB;

    const int tid  = threadIdx.x;
    const int lane = tid & 31;
    const int wave = tid >> 5;
    const int half = (tid & 16) ? 1 : 0;  // upper lane half: K rows 2,3 / M rows +8
    const int q16  = lane & 15;

    // ---- Async-stage xyz[b] (98304 B, 16B aligned) into LDS ----
    {
        const char* src = (const char*)(xyz + (size_t)b * BQ_N * 3);
        char* dst = (char*)raw3;
        const int nchunks = (BQ_N * 3 * 4) / 16;  // 6144
        for (int c = tid; c < nchunks; c += BQ_THREADS)
            async_copy_b128(src + (size_t)c * 16, dst + (size_t)c * 16);
    }
    wait_async0();
    __syncthreads();

    // ---- Per-wave query setup: A matrix and C=|q|^2 ----
    const int s_local = wave * 16 + q16;  // 0..63
    const int s_glob  = s_base + s_local;
    const size_t qbase = ((size_t)b * BQ_S + s_glob) * 3;
    const float qx = new_xyz[qbase + 0];
    const float qy = new_xyz[qbase + 1];
    const float qz = new_xyz[qbase + 2];
    const float qn = qx * qx + qy * qy + qz * qz;

    v2f amat;
    amat.x = half ? (-2.0f * qz) : (-2.0f * qx);  // K=2 : K=0
    amat.y = half ? (0.0f)       : (-2.0f * qy);  // K=3 : K=1

    v8f cinit;
#pragma unroll
    for (int v = 0; v < 8; ++v)
        cinit[v] = __shfl(qn, v + half * 8, 32);  // row v / v+8 -> |q_row|^2

    // ---- Scan points: WMMA distance tile + ordered first-32 selection ----
    int* myidx = idxlds + (wave * 16 + q16) * BQ_NS;  // owned by lanes 0..15
    int cnt = 0;

    for (int n0 = 0; n0 < BQ_N; n0 += 16) {
        const int   pi = n0 + q16;
        const float px = raw3[3 * pi + 0];
        const float py = raw3[3 * pi + 1];
        const float pz = raw3[3 * pi + 2];
        const float thr = BQ_R2 - (px * px + py * py + pz * pz);  // per-column threshold

        v2f bmat;
        bmat.x = half ? pz   : px;  // K=2 : K=0
        bmat.y = half ? 0.0f : py;  // K=3 : K=1

        v8f d = __builtin_amdgcn_wmma_f32_16x16x4_f32(
            false, amat, false, bmat, (short)0, cinit, false, false);

        const unsigned m0 = (unsigned)__ballot(d[0] <= thr);
        const unsigned m1 = (unsigned)__ballot(d[1] <= thr);
        const unsigned m2 = (unsigned)__ballot(d[2] <= thr);
        const unsigned m3 = (unsigned)__ballot(d[3] <= thr);
        const unsigned m4 = (unsigned)__ballot(d[4] <= thr);
        const unsigned m5 = (unsigned)__ballot(d[5] <= thr);
        const unsigned m6 = (unsigned)__ballot(d[6] <= thr);
        const unsigned m7 = (unsigned)__ballot(d[7] <= thr);

        const unsigned anym = m0 | m1 | m2 | m3 | m4 | m5 | m6 | m7;  // uniform
        if (anym) {
            if (lane < 16 && cnt < BQ_NS) {
                const int qs = lane & 7;
                // branchless select of this row's ballot word
                unsigned mm = m0;
                mm = (qs == 1) ? m1 : mm;
                mm = (qs == 2) ? m2 : mm;
                mm = (qs == 3) ? m3 : mm;
                mm = (qs == 4) ? m4 : mm;
                mm = (qs == 5) ? m5 : mm;
                mm = (qs == 6) ? m6 : mm;
                mm = (qs == 7) ? m7 : mm;
                // rows 0-7: bits [15:0] of m[row]; rows 8-15: bits [31:16] of m[row-8]
                unsigned mask16 = (lane & 8) ? (mm >> 16) : (mm & 0xFFFFu);
                while (mask16 && cnt < BQ_NS) {
                    const int j = __builtin_ctz(mask16);
                    mask16 &= mask16 - 1u;
                    myidx[cnt++] = n0 + j;
                }
            }
            const bool done = (lane < 16) ? (cnt >= BQ_NS) : true;
            if (__all(done)) break;
        }
    }

    // ---- Pad: repeat first index; empty ball -> clamp-to-last (JAX gather clamp) ----
    if (lane < 16) {
        const int pad = (cnt == 0) ? (BQ_N - 1) : myidx[0];
        for (int k = cnt; k < BQ_NS; ++k) myidx[k] = pad;
    }
    __syncthreads();

    // ---- Phase B: gather + write out[b][ch][s][k], coalesced in k ----
    for (int sq = 0; sq < BQ_QPB; ++sq) {
        const int s = s_base + sq;
        for (int e = tid; e < BQ_OUTC * BQ_NS; e += BQ_THREADS) {
            const int ch = e >> 5;
            const int k  = e & 31;
            const int n  = idxlds[sq * BQ_NS + k];
            float val;
            if (ch < 3) {
                val = raw3[3 * n + ch] - new_xyz[((size_t)b * BQ_S + s) * 3 + ch];
            } else {
                val = features[((size_t)b * BQ_C + (ch - 3)) * BQ_N + n];
            }
            out[(((size_t)b * BQ_OUTC + ch) * BQ_S + s) * BQ_NS + k] = val;
        }
    }
}

extern "C" void kernel_launch(void* const* d_in, const int* in_sizes, int n_in,
                              void* d_out, int out_size, void* d_ws, size_t ws_size,
                              hipStream_t stream) {
    (void)in_sizes; (void)n_in; (void)d_ws; (void)ws_size; (void)out_size;
    const float* xyz      = (const float*)d_in[0];
    const float* new_xyz  = (const float*)d_in[1];
    const float* features = (const float*)d_in[2];
    float* out = (float*)d_out;

    const size_t smem = sizeof(float) * BQ_N * 3 + sizeof(int) * BQ_QPB * BQ_NS; // 104KB
    (void)hipFuncSetAttribute((const void*)ball_query_group_kernel,
                              hipFuncAttributeMaxDynamicSharedMemorySize, (int)smem);

    const int grid = BQ_B * (BQ_S / BQ_QPB);  // 256 blocks
    ball_query_group_kernel<<<grid, BQ_THREADS, smem, stream>>>(xyz, new_xyz, features, out);
}